// LNet_xyt_Batch_71975061946472
// MI455X (gfx1250) — compile-verified
//
#include <hip/hip_runtime.h>
#include <hip/hip_bf16.h>

typedef __attribute__((ext_vector_type(2))) float v2f;
typedef __attribute__((ext_vector_type(8))) float v8f;

#define NBATCH 2048
#define NT 16
#define NPIX 1024  // 32*32

// One block = 4 waves (128 threads) per batch.
// Wave w owns columns [w*256, w*256+256) for both Gram and reconstruction.
__global__ __launch_bounds__(128) void lnet_svt_kernel(const float* __restrict__ in,
                                                       const float* __restrict__ thres_coef,
                                                       float* __restrict__ out)
{
    __shared__ float sGpart[4 * 256];  // per-wave partial Gram (also reused as X temp)
    __shared__ float sG[256];          // Gram / eigen-iterate
    __shared__ float sU[256];          // eigenvectors
    __shared__ float sM[256];          // M = U diag(w) U^T
    __shared__ float sW[16];           // S then w
    __shared__ float sRotC[8], sRotS[8];
    __shared__ int   sRotP[8], sRotQ[8];

    const int b    = blockIdx.x;
    const int tid  = threadIdx.x;
    const int lane = tid & 31;
    const int wave = tid >> 5;
    const int half = lane >> 4;   // 0: lanes 0-15, 1: lanes 16-31
    const int l16  = lane & 15;

    const float* A = in + (size_t)b * (NT * NPIX);
    const int colBase = wave * 256;

    // ---------------- Pass 1: partial Gram G_w = A_w * A_w^T via WMMA ----------------
    // For A*A^T the 16x4 A-operand layout equals the 4x16 B-operand layout:
    //   lane<16 : (A[l16][k+0], A[l16][k+1]);  lane>=16 : (A[l16][k+2], A[l16][k+3])
    v8f acc0 = {};
    v8f acc1 = {};
    for (int k = 0; k < 256; k += 8) {
        const float* p0 = A + (size_t)l16 * NPIX + (colBase + k + 2 * half);
        const float* p1 = p0 + 4;
        v2f ab0; ab0.x = p0[0]; ab0.y = p0[1];
        v2f ab1; ab1.x = p1[0]; ab1.y = p1[1];
        acc0 = __builtin_amdgcn_wmma_f32_16x16x4_f32(false, ab0, false, ab0, (short)0, acc0, false, false);
        acc1 = __builtin_amdgcn_wmma_f32_16x16x4_f32(false, ab1, false, ab1, (short)0, acc1, false, false);
    }
    {
        v8f acc = acc0 + acc1;
        float* gp = sGpart + wave * 256;
        #pragma unroll
        for (int r = 0; r < 8; ++r)
            gp[(half * 8 + r) * 16 + l16] = acc[r];   // C layout: VGPR r -> row half*8+r, col l16
    }
    __syncthreads();
    for (int i = tid; i < 256; i += 128)
        sG[i] = sGpart[i] + sGpart[256 + i] + sGpart[512 + i] + sGpart[768 + i];
    __syncthreads();

    // ---------------- Eigensolve (wave 0 only; barrier-free, same-wave LDS is in-order) --------
    if (wave == 0) {
        for (int i = lane; i < 256; i += 32)
            sU[i] = ((i >> 4) == (i & 15)) ? 1.0f : 0.0f;

        for (int sweep = 0; sweep < 7; ++sweep) {
            for (int r = 0; r < 15; ++r) {
                // 8 disjoint pairs via round-robin tournament; lanes 0..7 compute rotations
                if (lane < 8) {
                    int m  = lane;
                    int sa = (m == 0) ? 0 : ((m - 1 + r) % 15) + 1;   // player at slot m
                    int sb = ((14 - m + r) % 15) + 1;                 // player at slot 15-m
                    int p = sa < sb ? sa : sb;
                    int q = sa < sb ? sb : sa;
                    float app = sG[p * 16 + p];
                    float aqq = sG[q * 16 + q];
                    float apq = sG[p * 16 + q];
                    float c = 1.0f, s = 0.0f;
                    if (fabsf(apq) > 1e-20f) {
                        float tau = (aqq - app) / (2.0f * apq);
                        float t = copysignf(1.0f, tau) / (fabsf(tau) + sqrtf(1.0f + tau * tau));
                        c = rsqrtf(1.0f + t * t);
                        s = t * c;
                    }
                    sRotC[m] = c; sRotS[m] = s; sRotP[m] = p; sRotQ[m] = q;
                }
                // Row phase (J^T G): 32 lanes = (half selects pair parity) x (16 columns)
                #pragma unroll
                for (int mm = 0; mm < 4; ++mm) {
                    int m = 2 * mm + half;
                    int p = sRotP[m], q = sRotQ[m];
                    float c = sRotC[m], s = sRotS[m];
                    float gp = sG[p * 16 + l16], gq = sG[q * 16 + l16];
                    sG[p * 16 + l16] = c * gp - s * gq;
                    sG[q * 16 + l16] = s * gp + c * gq;
                }
                // Column phase (G J) + eigenvector update (U J)
                #pragma unroll
                for (int mm = 0; mm < 4; ++mm) {
                    int m = 2 * mm + half;
                    int p = sRotP[m], q = sRotQ[m];
                    float c = sRotC[m], s = sRotS[m];
                    float gp = sG[l16 * 16 + p], gq = sG[l16 * 16 + q];
                    sG[l16 * 16 + p] = c * gp - s * gq;
                    sG[l16 * 16 + q] = s * gp + c * gq;
                    float up = sU[l16 * 16 + p], uq = sU[l16 * 16 + q];
                    sU[l16 * 16 + p] = c * up - s * uq;
                    sU[l16 * 16 + q] = s * up + c * uq;
                }
            }
        }

        // Singular values, learned soft-threshold, spectral weights w = S_new / S
        if (lane < 16) {
            float lam = sG[lane * 16 + lane];
            sW[lane] = sqrtf(fmaxf(lam, 0.0f));          // S
        }
        if (lane < 16) {
            float smax = 0.0f;
            for (int k = 0; k < 16; ++k) smax = fmaxf(smax, sW[k]);
            float tc    = thres_coef[b];
            float thres = smax / (1.0f + __expf(-tc));   // sigmoid(tc) * S_max
            float S     = sW[lane];
            float d     = S - thres;
            float Snew  = fmaxf(d, 0.0f) + thres / (1.0f + __expf(-d));
            sW[lane]    = Snew / fmaxf(S, 1e-12f);
        }

        // X = U diag(w) (into sGpart), then M = X U^T
        for (int e = lane; e < 256; e += 32)
            sGpart[e] = sU[e] * sW[e & 15];
        for (int e = lane; e < 256; e += 32) {
            int i = e >> 4, j = e & 15;
            float a = 0.0f;
            #pragma unroll
            for (int k = 0; k < 16; ++k)
                a += sGpart[i * 16 + k] * sU[j * 16 + k];
            sM[e] = a;
        }
    }
    __syncthreads();

    // ---------------- Pass 2: L = M * A via WMMA (16x16x16 per tile = 4 chained k4) --------
    // Preload A-operand slabs of M: chunk c -> (M[l16][4c+2*half], M[l16][4c+2*half+1])
    v2f m0, m1, m2, m3;
    {
        const float* Mp = sM + l16 * 16 + 2 * half;
        m0.x = Mp[0];  m0.y = Mp[1];
        m1.x = Mp[4];  m1.y = Mp[5];
        m2.x = Mp[8];  m2.y = Mp[9];
        m3.x = Mp[12]; m3.y = Mp[13];
    }
    float* O = out + (size_t)b * (NT * NPIX);
    for (int tile = 0; tile < 16; ++tile) {
        const int cb = colBase + tile * 16;
        v8f d = {};
        #pragma unroll
        for (int c = 0; c < 4; ++c) {
            // B operand (4x16 slab of A): VGPR0 = row 4c+2*half, VGPR1 = row 4c+2*half+1
            const float* p = A + (size_t)(4 * c + 2 * half) * NPIX + cb + l16;
            v2f bb; bb.x = p[0]; bb.y = p[NPIX];
            v2f ma = (c == 0) ? m0 : (c == 1) ? m1 : (c == 2) ? m2 : m3;
            d = __builtin_amdgcn_wmma_f32_16x16x4_f32(false, ma, false, bb, (short)0, d, false, false);
        }
        #pragma unroll
        for (int r = 0; r < 8; ++r)
            O[(size_t)(half * 8 + r) * NPIX + cb + l16] = d[r];  // two 64B segments per store
    }
}

extern "C" void kernel_launch(void* const* d_in, const int* in_sizes, int n_in,
                              void* d_out, int out_size, void* d_ws, size_t ws_size,
                              hipStream_t stream) {
    const float* in  = (const float*)d_in[0];   // (2048, 16, 32, 32) f32
    const float* tc  = (const float*)d_in[1];   // (2048,) f32
    float* out = (float*)d_out;                 // (2048, 16, 32, 32) f32
    (void)in_sizes; (void)n_in; (void)out_size; (void)d_ws; (void)ws_size;
    lnet_svt_kernel<<<dim3(NBATCH), dim3(128), 0, stream>>>(in, tc, out);
}